// UltraTinyODWithPost_27058293965494
// MI455X (gfx1250) — compile-verified
//
#include <hip/hip_runtime.h>
#include <stdint.h>

// ---------------- static problem config (matches reference) ----------------
#define NB    16          // batch
#define NANC  3           // anchors
#define NCLS  80          // classes
#define NOCH  86          // 6 head + 80 classes
#define IH    160
#define IW    160
#define HW    (IH*IW)     // 25600
#define NLOC  (NANC*HW)   // 76800 per batch
#define K_TOP 100
#define CONF  0.25f
#define NBKT  1024        // float-bit buckets (bits>>20, monotonic for s in [0,1])
#define CAP   4096        // candidate cap per batch (region A: 128, region B: CAP-128)
#define CAPA  128

// ---------------- CDNA5 async global->LDS helpers ----------------
typedef __attribute__((address_space(1))) int* as1_ip;
typedef __attribute__((address_space(3))) int* as3_ip;

__device__ __forceinline__ void async_ld_f32(const float* g, float* l) {
#if __has_builtin(__builtin_amdgcn_global_load_async_to_lds_b32)
  __builtin_amdgcn_global_load_async_to_lds_b32(
      (as1_ip)(uintptr_t)g, (as3_ip)(uint32_t)(uintptr_t)l, 0, 0);
#else
  uint32_t loff = (uint32_t)(uintptr_t)l;   // low 32 bits of generic LDS addr = LDS offset
  asm volatile("global_load_async_to_lds_b32 %0, %1, off"
               :: "v"(loff), "v"(g) : "memory");
#endif
}

#if defined(__HIP_DEVICE_COMPILE__) && __has_builtin(__builtin_amdgcn_s_wait_asynccnt)
#define WAIT_ASYNC(n) do { __builtin_amdgcn_s_wait_asynccnt(n); asm volatile("" ::: "memory"); } while (0)
#else
#define WAIT_ASYNC(n) asm volatile("s_wait_asynccnt " #n ::: "memory")
#endif

// ---------------- math helpers ----------------
__device__ __forceinline__ float sigm(float x) { return 1.0f / (1.0f + __expf(-x)); }
__device__ __forceinline__ float softplusf(float x) {
  return fmaxf(x, 0.0f) + log1pf(__expf(-fabsf(x)));   // stable, matches jax.nn.softplus
}

// ---------------- stage 1: per-location thresholded best score ----------------
// grid (HW/256, NANC, NB), block 256.  Class logits stream through LDS via
// double-buffered async-to-LDS copies (4 chunks x 20 channels).  Each lane only
// reads the LDS column it filled itself, so no workgroup barriers are needed.
__global__ __launch_bounds__(256)
void k_score(const float* __restrict__ raw, float* __restrict__ scores) {
  __shared__ float lds[2][20 * 256];           // 40 KB
  const int tid  = threadIdx.x;
  const int tile = blockIdx.x;
  const int a    = blockIdx.y;
  const int b    = blockIdx.z;
  const int pos  = tile * 256 + tid;
  const float* base = raw + ((size_t)(b * NANC + a) * NOCH) * HW + pos;

  auto issue = [&](int q, int buf) {
    const float* g = base + (size_t)(6 + q * 20) * HW;
#pragma unroll
    for (int c = 0; c < 20; ++c)
      async_ld_f32(g + (size_t)c * HW, &lds[buf][c * 256 + tid]);
  };
  issue(0, 0);
  issue(1, 1);

  // head channels overlap the async stream (only obj/quality needed here)
  float obj_l = base[(size_t)4 * HW];
  float qua_l = base[(size_t)5 * HW];

  float m = -3.4e38f;
  WAIT_ASYNC(20);                               // chunk0 done (in-order completion)
#pragma unroll
  for (int c = 0; c < 20; ++c) m = fmaxf(m, lds[0][c * 256 + tid]);
  issue(2, 0);
  WAIT_ASYNC(20);                               // chunk1 done
#pragma unroll
  for (int c = 0; c < 20; ++c) m = fmaxf(m, lds[1][c * 256 + tid]);
  issue(3, 1);
  WAIT_ASYNC(20);                               // chunk2 done
#pragma unroll
  for (int c = 0; c < 20; ++c) m = fmaxf(m, lds[0][c * 256 + tid]);
  WAIT_ASYNC(0);                                // chunk3 done
#pragma unroll
  for (int c = 0; c < 20; ++c) m = fmaxf(m, lds[1][c * 256 + tid]);

  // max_c sigma(l_c) == sigma(max_c l_c)  (sigmoid monotone)
  float s = sigm(obj_l) * sigm(qua_l) * sigm(m);
  s = (s >= CONF) ? s : 0.0f;
  scores[(size_t)b * NLOC + a * HW + pos] = s;
}

// ---------------- stage 2a: zero hist + counters ----------------
__global__ void k_zero(unsigned* __restrict__ hist_cnt) {
  int g = blockIdx.x * blockDim.x + threadIdx.x;
  if (g < NB * NBKT + 2 * NB) hist_cnt[g] = 0u;
}

// ---------------- stage 2b: per-batch float-bit histogram ----------------
__global__ __launch_bounds__(1024)
void k_hist(const float* __restrict__ scores, unsigned* __restrict__ hist) {
  __shared__ unsigned lh[NBKT];
  const int tid = threadIdx.x;
  const int b   = blockIdx.y;
  lh[tid] = 0u;
  __syncthreads();
  const int idx = blockIdx.x * 1024 + tid;     // < 76800 by construction
  float s = scores[(size_t)b * NLOC + idx];
  atomicAdd(&lh[__float_as_uint(s) >> 20], 1u);
  __syncthreads();
  unsigned c = lh[tid];
  if (c) atomicAdd(&hist[b * NBKT + tid], c);
}

// ---------------- stage 3: find 100th-score bucket ----------------
__global__ void k_select(const unsigned* __restrict__ hist, unsigned* __restrict__ thr) {
  const int b = threadIdx.x;
  if (b >= NB) return;
  unsigned cum = 0, t = 0;
  for (int k = NBKT - 1; k >= 0; --k) {
    cum += hist[b * NBKT + k];
    if (cum >= K_TOP) { t = (unsigned)k; break; }
  }
  thr[b] = t;
}

// ---------------- stage 4: two-tier candidate compaction ----------------
__global__ __launch_bounds__(1024)
void k_compact(const float* __restrict__ scores, const unsigned* __restrict__ thr,
               unsigned* __restrict__ cnt, unsigned long long* __restrict__ cand) {
  const int b   = blockIdx.y;
  const int idx = blockIdx.x * 1024 + threadIdx.x;
  float s = scores[(size_t)b * NLOC + idx];
  unsigned bits = __float_as_uint(s);
  unsigned bkt  = bits >> 20;
  unsigned t    = thr[b];
  // key: score desc, then smaller index first (~idx larger <=> idx smaller)
  unsigned long long key = ((unsigned long long)bits << 32) | (unsigned)(~(unsigned)idx);
  if (bkt > t) {                                // strictly above: < K_TOP elems guaranteed
    unsigned slot = atomicAdd(&cnt[2 * b], 1u);
    if (slot < CAPA) cand[(size_t)b * CAP + slot] = key;
  } else if (bkt == t) {                        // boundary bucket: capped
    unsigned slot = atomicAdd(&cnt[2 * b + 1], 1u);
    if (slot < (CAP - CAPA)) cand[(size_t)b * CAP + CAPA + slot] = key;
  }
}

// ---------------- stage 5: per-batch bitonic sort (LDS) + emit ----------------
__global__ __launch_bounds__(1024)
void k_sortemit(const float* __restrict__ raw, const float* __restrict__ anchors,
                const unsigned* __restrict__ cnt, const unsigned long long* __restrict__ cand,
                float* __restrict__ out) {
  __shared__ unsigned long long key[CAP];      // 32 KB
  const int tid = threadIdx.x;
  const int b   = blockIdx.x;
  unsigned mA = cnt[2 * b];     if (mA > CAPA)       mA = CAPA;
  unsigned mB = cnt[2 * b + 1]; if (mB > CAP - CAPA) mB = CAP - CAPA;
  for (int i = tid; i < CAP; i += 1024) {
    unsigned long long v = 0ull;
    if (i < CAPA) { if ((unsigned)i < mA) v = cand[(size_t)b * CAP + i]; }
    else          { if ((unsigned)(i - CAPA) < mB) v = cand[(size_t)b * CAP + i]; }
    key[i] = v;
  }
  __syncthreads();
  for (unsigned kk = 2; kk <= CAP; kk <<= 1) {
    for (unsigned j = kk >> 1; j > 0; j >>= 1) {
      for (unsigned i = (unsigned)tid; i < CAP; i += 1024) {
        unsigned ixj = i ^ j;
        if (ixj > i) {
          unsigned long long A = key[i], Bv = key[ixj];
          bool desc = ((i & kk) == 0);
          if (desc ? (A < Bv) : (A > Bv)) { key[i] = Bv; key[ixj] = A; }
        }
      }
      __syncthreads();
    }
  }
  if (tid < K_TOP) {
    unsigned long long kv = key[tid];
    float s = __uint_as_float((unsigned)(kv >> 32));
    unsigned idx = ~(unsigned)kv;
    float* o = out + ((size_t)b * K_TOP + tid) * 6;
    if (idx >= (unsigned)NLOC) {               // defensive: should not happen (>=100 cands)
      o[0] = o[1] = o[2] = o[3] = o[4] = o[5] = 0.0f;
      return;
    }
    unsigned a = idx / HW, pos = idx % HW;
    unsigned y = pos / IW, x = pos % IW;
    const float* base = raw + ((size_t)(b * NANC + a) * NOCH) * HW + pos;
    float txv = base[0];
    float tyv = base[(size_t)HW];
    float twv = base[(size_t)2 * HW];
    float thv = base[(size_t)3 * HW];
    int cls = 0;
    if (s > 0.0f) {                            // argmax of logits == argmax of scores
      float best = -3.4e38f;
      for (int c = 0; c < NCLS; ++c) {
        float v = base[(size_t)(6 + c) * HW];
        if (v > best) { best = v; cls = c; }
      }
    }
    float cx = (sigm(txv) + (float)x) * (1.0f / IW);
    float cy = (sigm(tyv) + (float)y) * (1.0f / IH);
    float bw = anchors[2 * a + 0] * softplusf(twv);
    float bh = anchors[2 * a + 1] * softplusf(thv);
    o[0] = s; o[1] = (float)cls; o[2] = cx; o[3] = cy; o[4] = bw; o[5] = bh;
  }
}

// ---------------- launch ----------------
extern "C" void kernel_launch(void* const* d_in, const int* in_sizes, int n_in,
                              void* d_out, int out_size, void* d_ws, size_t ws_size,
                              hipStream_t stream) {
  (void)in_sizes; (void)n_in; (void)out_size; (void)ws_size;
  const float* raw     = (const float*)d_in[0];   // (16, 258, 160, 160) f32
  const float* anchors = (const float*)d_in[1];   // (3, 2) f32
  float* out = (float*)d_out;                     // (16, 100, 6) f32

  // workspace layout (total ~5.5 MB)
  char* ws = (char*)d_ws;
  size_t off = 0;
  float* scores = (float*)(ws + off);                 off += (size_t)NB * NLOC * sizeof(float); // 4,915,200
  unsigned long long* cand = (unsigned long long*)(ws + off); off += (size_t)NB * CAP * 8;      //   524,288
  unsigned* hist = (unsigned*)(ws + off);             off += (size_t)NB * NBKT * 4;             //    65,536
  unsigned* cnt  = (unsigned*)(ws + off);             off += (size_t)2 * NB * 4;                // adjacent to hist (k_zero covers both)
  unsigned* thr  = (unsigned*)(ws + off);             off += (size_t)NB * 4;

  k_zero   <<<(NB * NBKT + 2 * NB + 255) / 256, 256, 0, stream>>>(hist);
  k_score  <<<dim3(HW / 256, NANC, NB), 256, 0, stream>>>(raw, scores);
  k_hist   <<<dim3(NLOC / 1024, NB), 1024, 0, stream>>>(scores, hist);
  k_select <<<1, 32, 0, stream>>>(hist, thr);
  k_compact<<<dim3(NLOC / 1024, NB), 1024, 0, stream>>>(scores, thr, cnt, cand);
  k_sortemit<<<NB, 1024, 0, stream>>>(raw, anchors, cnt, cand, out);
}